// PiecewiseWallCollisionIntervention_9363028706206
// MI455X (gfx1250) — compile-verified
//
#include <hip/hip_runtime.h>

// Problem constants (from reference)
#define BN_NODES 16
#define BN_WALLS 32
#define WALL_ROW_F 256   // floats per batch in wall0/wall1: 32 walls * 2 ends * 4 intervals

// One wave32 per batch; lane = wall index. Memory-bound kernel (~224 MB @ 23.3 TB/s).
__global__ __launch_bounds__(256) void wall_collision_kernel(
    const float* __restrict__ y,       // (B, 16)
    const float* __restrict__ y_prev,  // (B, 16)
    const float* __restrict__ wall0,   // (B, 32, 2, 4)  x-coords
    const float* __restrict__ wall1,   // (B, 32, 2, 4)  y-coords
    const int*   __restrict__ interval_idx_p, // scalar
    float*       __restrict__ out,     // (B, 16)
    int B, int wave_stride)
{
    const int tid  = blockIdx.x * blockDim.x + threadIdx.x;
    const int wave = tid >> 5;
    const int lane = tid & 31;
    const int iidx = interval_idx_p[0];   // uniform -> scalar load

    for (int b = wave; b < B; b += wave_stride) {
        // Prefetch next iteration's wall stream (one-touch data -> low locality hint)
        const int bn = b + wave_stride;
        if (bn < B) {
            __builtin_prefetch(wall0 + (size_t)bn * WALL_ROW_F + lane * 8, 0, 0);
            __builtin_prefetch(wall1 + (size_t)bn * WALL_ROW_F + lane * 8, 0, 0);
        }

        // Lane w reads wall w: offsets w*8 + {0,4} + interval within the 256-float batch row.
        const size_t wbase = (size_t)b * WALL_ROW_F + (size_t)lane * 8 + (size_t)iidx;
        const float w0x = wall0[wbase];
        const float w1x = wall0[wbase + 4];
        const float w0y = wall1[wbase];
        const float w1y = wall1[wbase + 4];

        // p0 = y_prev[b, 0:2] (uniform per wave, one line fetch)
        const float2 p0v = *(const float2*)(y_prev + (size_t)b * BN_NODES);
        const float p0x = p0v.x, p0y = p0v.y;

        // Stage the y row for copy-through: lanes 0..15 hold y[b, lane]
        const float yv = y[(size_t)b * BN_NODES + (lane & 15)];
        const float p1x = __shfl(yv, 0, 32);
        const float p1y = __shfl(yv, 1, 32);

        // Segment-segment intersection (per-lane = per-wall)
        const float dx = p1x - p0x, dy = p1y - p0y;
        const float ex = w1x - w0x, ey = w1y - w0y;
        const float fx = w0x - p0x, fy = w0y - p0y;
        const float denom = dx * ey - dy * ex;
        const float denom_safe = (denom == 0.0f) ? 1.0f : denom;
        const float t = (fx * ey - fy * ex) / denom_safe;
        const float u = (fx * dy - fy * dx) / denom_safe;
        const bool valid = (denom != 0.0f) &&
                           (t >= 0.0f) && (t <= 1.0f) &&
                           (u >= 0.0f) && (u <= 1.0f);

        // argmin key: valid -> t in [0,1]; invalid -> +inf (matches where(isnan, inf, t_coll))
        float key = valid ? t : __builtin_inff();
        float rx  = valid ? (p0x + t * dx) : p1x;
        float ry  = valid ? (p0y + t * dy) : p1y;
        int   widx = lane;

        // Wave32 butterfly min-reduction with first-index tie-break (jnp.argmin semantics)
        #pragma unroll
        for (int off = 16; off; off >>= 1) {
            const float okey = __shfl_xor(key,  off, 32);
            const float orx  = __shfl_xor(rx,   off, 32);
            const float ory  = __shfl_xor(ry,   off, 32);
            const int   oidx = __shfl_xor(widx, off, 32);
            const bool take = (okey < key) || ((okey == key) && (oidx < widx));
            key  = take ? okey : key;
            rx   = take ? orx  : rx;
            ry   = take ? ory  : ry;
            widx = take ? oidx : widx;
        }

        // Write output row: out[b] = y[b] with [0:2] = hit point
        if (lane < BN_NODES) {
            const float v = (lane == 0) ? rx : ((lane == 1) ? ry : yv);
            out[(size_t)b * BN_NODES + lane] = v;
        }
    }
}

extern "C" void kernel_launch(void* const* d_in, const int* in_sizes, int n_in,
                              void* d_out, int out_size, void* d_ws, size_t ws_size,
                              hipStream_t stream) {
    const float* y     = (const float*)d_in[0];
    const float* yprev = (const float*)d_in[1];
    const float* wall0 = (const float*)d_in[2];
    const float* wall1 = (const float*)d_in[3];
    const int*   iidx  = (const int*)d_in[4];
    float* out = (float*)d_out;

    const int B = in_sizes[0] / BN_NODES;

    // ~4 batches per wave: strided loop enables prefetch of the streaming wall data,
    // while 25k waves still vastly oversubscribe the WGPs for latency hiding.
    const int BATCHES_PER_WAVE = 4;
    const int waves_wanted = (B + BATCHES_PER_WAVE - 1) / BATCHES_PER_WAVE;
    const int threads = 256;                       // 8 wave32 per workgroup
    const int blocks = (waves_wanted * 32 + threads - 1) / threads;
    const int wave_stride = blocks * (threads / 32);

    wall_collision_kernel<<<blocks, threads, 0, stream>>>(
        y, yprev, wall0, wall1, iidx, out, B, wave_stride);
}